// WaveletTransform_46050639348243
// MI455X (gfx1250) — compile-verified
//
#include <hip/hip_runtime.h>

typedef float f4 __attribute__((ext_vector_type(4)));
typedef float f2 __attribute__((ext_vector_type(2)));

__device__ __forceinline__ void haar4(float a, float b, float c, float d,
                                      float& ll, float& lh, float& hl, float& hh) {
    ll = 0.5f * (a + b + c + d);
    lh = 0.5f * (a - b + c - d);
    hl = 0.5f * (a + b - c - d);
    hh = 0.5f * (a - b - c + d);
}

// ---------------------------------------------------------------------------
// Level-1 subbands of x channels 8..31 -> high0 [4,96,256,256]
// Thread: 4 output pixels of one row (8 input cols x 2 rows).
// Grid: 4*24*256*64 threads = 6144 blocks x 256. Pure streaming, NT both ways.
// ---------------------------------------------------------------------------
__global__ __launch_bounds__(256) void haar_l1_high(const float* __restrict__ x,
                                                    float* __restrict__ high0) {
    int idx = blockIdx.x * 256 + threadIdx.x;
    int x8  = idx & 63;           // 8-col strip index (64 per row)
    int y   = (idx >> 6) & 255;   // output row
    int t   = idx >> 14;          // b*24 + c24
    int c24 = t % 24;
    int b   = t / 24;

    const float* p = x + (((b * 32 + 8 + c24) * 512 + 2 * y) * 512 + 8 * x8);
    f4 r0a = __builtin_nontemporal_load((const f4*)p);
    f4 r0b = __builtin_nontemporal_load((const f4*)(p + 4));
    f4 r1a = __builtin_nontemporal_load((const f4*)(p + 512));
    f4 r1b = __builtin_nontemporal_load((const f4*)(p + 516));

    float ll[4], lh[4], hl[4], hh[4];
    haar4(r0a.x, r0a.y, r1a.x, r1a.y, ll[0], lh[0], hl[0], hh[0]);
    haar4(r0a.z, r0a.w, r1a.z, r1a.w, ll[1], lh[1], hl[1], hh[1]);
    haar4(r0b.x, r0b.y, r1b.x, r1b.y, ll[2], lh[2], hl[2], hh[2]);
    haar4(r0b.z, r0b.w, r1b.z, r1b.w, ll[3], lh[3], hl[3], hh[3]);

    f4 vll = {ll[0], ll[1], ll[2], ll[3]};
    f4 vlh = {lh[0], lh[1], lh[2], lh[3]};
    f4 vhl = {hl[0], hl[1], hl[2], hl[3]};
    f4 vhh = {hh[0], hh[1], hh[2], hh[3]};

    const int ps = 256 * 256;
    float* o = high0 + (((b * 96 + 4 * c24) * 256 + y) * 256 + 4 * x8);
    __builtin_nontemporal_store(vll, (f4*)(o));
    __builtin_nontemporal_store(vlh, (f4*)(o + ps));
    __builtin_nontemporal_store(vhl, (f4*)(o + 2 * ps));
    __builtin_nontemporal_store(vhh, (f4*)(o + 3 * ps));
}

// ---------------------------------------------------------------------------
// Two fused Haar levels of x channels 2..7 -> high1 [4,96,128,128]
// Output channel = 16*c + 4*s0 + s1 - 32. Thread: 2 output pixels (8x4 input).
// Grid: 4*6*128*64 threads = 768 blocks x 256.
// ---------------------------------------------------------------------------
__global__ __launch_bounds__(256) void haar_l2_high(const float* __restrict__ x,
                                                    float* __restrict__ high1) {
    int idx = blockIdx.x * 256 + threadIdx.x;
    int xo = idx & 63;            // output col pair index (64 per row)
    int y  = (idx >> 6) & 127;    // output row
    int t  = idx >> 13;           // b*6 + c6
    int c6 = t % 6;
    int b  = t / 6;

    const float* p = x + (((b * 32 + 2 + c6) * 512 + 4 * y) * 512 + 8 * xo);
    f4 r[4][2];
#pragma unroll
    for (int i = 0; i < 4; ++i) {
        r[i][0] = __builtin_nontemporal_load((const f4*)(p + 512 * i));
        r[i][1] = __builtin_nontemporal_load((const f4*)(p + 512 * i + 4));
    }

    float out2[4][4][2];          // [s0][s1][px]
#pragma unroll
    for (int px = 0; px < 2; ++px) {
        float L1[4][2][2];        // level-1 subbands, 2x2 field
#pragma unroll
        for (int yy = 0; yy < 2; ++yy)
#pragma unroll
            for (int xx = 0; xx < 2; ++xx)
                haar4(r[2 * yy][px][2 * xx],     r[2 * yy][px][2 * xx + 1],
                      r[2 * yy + 1][px][2 * xx], r[2 * yy + 1][px][2 * xx + 1],
                      L1[0][yy][xx], L1[1][yy][xx], L1[2][yy][xx], L1[3][yy][xx]);
#pragma unroll
        for (int s0 = 0; s0 < 4; ++s0)
            haar4(L1[s0][0][0], L1[s0][0][1], L1[s0][1][0], L1[s0][1][1],
                  out2[s0][0][px], out2[s0][1][px], out2[s0][2][px], out2[s0][3][px]);
    }

    const int ps = 128 * 128;
    float* base = high1 + ((b * 96) * 128 + y) * 128 + 2 * xo;
#pragma unroll
    for (int s0 = 0; s0 < 4; ++s0)
#pragma unroll
        for (int s1 = 0; s1 < 4; ++s1) {
            int ch = 16 * (c6 + 2) + 4 * s0 + s1 - 32;   // 0..95
            f2 v = {out2[s0][s1][0], out2[s0][s1][1]};
            __builtin_nontemporal_store(v, (f2*)(base + ch * ps));
        }
}

// ---------------------------------------------------------------------------
// Three fused Haar levels of x channels 0..1 -> high2 [4,96,64,64] and
// low2 [4,32,64,64] written twice (output slots 0 and 4).
// c2 = 16*cx + 4*s0 + s1 (low1 channel). c2>=8 -> high2 ch 4*c2+s2-32;
// c2<8 -> low2 ch 4*c2+s2. Thread: one 8x8 input block.
// Grid: 4*2*64*64 threads = 128 blocks x 256.
// ---------------------------------------------------------------------------
__global__ __launch_bounds__(256) void haar_l3(const float* __restrict__ x,
                                               float* __restrict__ low2a,
                                               float* __restrict__ high2,
                                               float* __restrict__ low2b) {
    int idx = blockIdx.x * 256 + threadIdx.x;
    int xo = idx & 63;
    int y  = (idx >> 6) & 63;
    int t  = idx >> 12;           // b*2 + cx
    int cx = t & 1;
    int b  = t >> 1;

    const float* p = x + (((b * 32 + cx) * 512 + 8 * y) * 512 + 8 * xo);
    float v[8][8];
#pragma unroll
    for (int i = 0; i < 8; ++i) {
        f4 lo = __builtin_nontemporal_load((const f4*)(p + 512 * i));
        f4 hi = __builtin_nontemporal_load((const f4*)(p + 512 * i + 4));
        v[i][0] = lo.x; v[i][1] = lo.y; v[i][2] = lo.z; v[i][3] = lo.w;
        v[i][4] = hi.x; v[i][5] = hi.y; v[i][6] = hi.z; v[i][7] = hi.w;
    }

    float L1[4][4][4];            // [s0][y][x] level-1, 4x4 field
#pragma unroll
    for (int yy = 0; yy < 4; ++yy)
#pragma unroll
        for (int xx = 0; xx < 4; ++xx)
            haar4(v[2 * yy][2 * xx],     v[2 * yy][2 * xx + 1],
                  v[2 * yy + 1][2 * xx], v[2 * yy + 1][2 * xx + 1],
                  L1[0][yy][xx], L1[1][yy][xx], L1[2][yy][xx], L1[3][yy][xx]);

    float L2[4][4][2][2];         // [s0][s1][y][x] level-2, 2x2 field
#pragma unroll
    for (int s0 = 0; s0 < 4; ++s0)
#pragma unroll
        for (int yy = 0; yy < 2; ++yy)
#pragma unroll
            for (int xx = 0; xx < 2; ++xx)
                haar4(L1[s0][2 * yy][2 * xx],     L1[s0][2 * yy][2 * xx + 1],
                      L1[s0][2 * yy + 1][2 * xx], L1[s0][2 * yy + 1][2 * xx + 1],
                      L2[s0][0][yy][xx], L2[s0][1][yy][xx],
                      L2[s0][2][yy][xx], L2[s0][3][yy][xx]);

    const int ps = 64 * 64;
    int pix = y * 64 + xo;
#pragma unroll
    for (int s0 = 0; s0 < 4; ++s0)
#pragma unroll
        for (int s1 = 0; s1 < 4; ++s1) {
            float l3[4];
            haar4(L2[s0][s1][0][0], L2[s0][s1][0][1],
                  L2[s0][s1][1][0], L2[s0][s1][1][1],
                  l3[0], l3[1], l3[2], l3[3]);
            int c2 = 16 * cx + 4 * s0 + s1;   // low1 channel 0..31 (wave-uniform branch)
#pragma unroll
            for (int s2 = 0; s2 < 4; ++s2) {
                if (c2 >= 8) {
                    int ch = 4 * c2 + s2 - 32;                 // 0..95
                    __builtin_nontemporal_store(l3[s2], high2 + (b * 96 + ch) * ps + pix);
                } else {
                    int ch = 4 * c2 + s2;                      // 0..31
                    __builtin_nontemporal_store(l3[s2], low2a + (b * 32 + ch) * ps + pix);
                    __builtin_nontemporal_store(l3[s2], low2b + (b * 32 + ch) * ps + pix);
                }
            }
        }
}

extern "C" void kernel_launch(void* const* d_in, const int* in_sizes, int n_in,
                              void* d_out, int out_size, void* d_ws, size_t ws_size,
                              hipStream_t stream) {
    (void)in_sizes; (void)n_in; (void)d_ws; (void)ws_size; (void)out_size;
    const float* x = (const float*)d_in[0];
    float* out = (float*)d_out;

    // Output layout (flat float offsets), return order (low, h0, h1, h2, low):
    float* low2a = out;                    //   524288  [4,32,64,64]
    float* high0 = out + 524288;           // 25165824  [4,96,256,256]
    float* high1 = out + 25690112;         //  6291456  [4,96,128,128]
    float* high2 = out + 31981568;         //  1572864  [4,96,64,64]
    float* low2b = out + 33554432;         //   524288  [4,32,64,64]

    haar_l1_high<<<6144, 256, 0, stream>>>(x, high0);         // 4*24*256*64 thr
    haar_l2_high<<<768, 256, 0, stream>>>(x, high1);          // 4*6*128*64  thr
    haar_l3    <<<128, 256, 0, stream>>>(x, low2a, high2, low2b); // 4*2*64*64 thr
}